// GaussianModel_85779086836034
// MI455X (gfx1250) — compile-verified
//
#include <hip/hip_runtime.h>

typedef float v4f __attribute__((ext_vector_type(4)));
typedef float v2f __attribute__((ext_vector_type(2)));

// gcc-style vector to match the async-LDS builtin's parameter type exactly
typedef int v4i __attribute__((vector_size(16)));

#define PTS 256  // points per block; 256*3 floats staged per stride-3 array

#if defined(__gfx1250__) && __has_builtin(__builtin_amdgcn_global_load_async_to_lds_b128)
#define HAVE_ASYNC_LDS 1
typedef __attribute__((address_space(1))) v4i as1_v4i;  // global
typedef __attribute__((address_space(3))) v4i as3_v4i;  // LDS
#else
#define HAVE_ASYNC_LDS 0
#endif

// ---- per-point math + streaming stores, shared by both kernels ----
__device__ __forceinline__ void gs_compute_store(
    int i,
    float px, float py, float pz,
    float l0, float l1, float l2,
    v4f q,
    const float* __restrict__ R_cam,
    const float* __restrict__ t_cam,
    const float* __restrict__ intr,
    float* __restrict__ out_means,
    float* __restrict__ out_sigma,
    float* __restrict__ out_depth)
{
    // uniform camera constants -> scalar loads
    float rc[9];
#pragma unroll
    for (int k = 0; k < 9; ++k) rc[k] = R_cam[k];
    const float tc0 = t_cam[0], tc1 = t_cam[1], tc2 = t_cam[2];
    const float fx = intr[0], fy = intr[1], cx = intr[2], cy = intr[3];

    // quaternion -> rotation matrix
    float w = q.x, x = q.y, y = q.z, z = q.w;
    const float rn = rsqrtf(w * w + x * x + y * y + z * z);
    w *= rn; x *= rn; y *= rn; z *= rn;

    const float r00 = 1.f - 2.f * (y * y + z * z);
    const float r01 = 2.f * (x * y - w * z);
    const float r02 = 2.f * (x * z + w * y);
    const float r10 = 2.f * (x * y + w * z);
    const float r11 = 1.f - 2.f * (x * x + z * z);
    const float r12 = 2.f * (y * z - w * x);
    const float r20 = 2.f * (x * z - w * y);
    const float r21 = 2.f * (y * z + w * x);
    const float r22 = 1.f - 2.f * (x * x + y * y);

    const float sx = __expf(l0);
    const float sy = __expf(l1);
    const float sz = __expf(l2);

    // RS = R * diag(s)
    const float a00 = r00 * sx, a01 = r01 * sy, a02 = r02 * sz;
    const float a10 = r10 * sx, a11 = r11 * sy, a12 = r12 * sz;
    const float a20 = r20 * sx, a21 = r21 * sy, a22 = r22 * sz;

    // Sigma3d = RS * RS^T (symmetric)
    const float c00 = a00 * a00 + a01 * a01 + a02 * a02;
    const float c01 = a00 * a10 + a01 * a11 + a02 * a12;
    const float c02 = a00 * a20 + a01 * a21 + a02 * a22;
    const float c11 = a10 * a10 + a11 * a11 + a12 * a12;
    const float c12 = a10 * a20 + a11 * a21 + a12 * a22;
    const float c22 = a20 * a20 + a21 * a21 + a22 * a22;

    // camera transform
    const float xc = rc[0] * px + rc[1] * py + rc[2] * pz + tc0;
    const float yc = rc[3] * px + rc[4] * py + rc[5] * pz + tc1;
    const float zc = rc[6] * px + rc[7] * py + rc[8] * pz + tc2;

    const float d    = fmaxf(zc, 0.01f);
    const float invd = 1.0f / d;

    const float mx = fmaf(xc * fx, invd, cx);
    const float my = fmaf(yc * fy, invd, cy);

    // JW = J @ R_cam
    const float fdx = fx * invd, fdy = fy * invd;
    const float gx = -fx * xc * invd * invd;
    const float gy = -fy * yc * invd * invd;
    const float jw00 = fdx * rc[0] + gx * rc[6];
    const float jw01 = fdx * rc[1] + gx * rc[7];
    const float jw02 = fdx * rc[2] + gx * rc[8];
    const float jw10 = fdy * rc[3] + gy * rc[6];
    const float jw11 = fdy * rc[4] + gy * rc[7];
    const float jw12 = fdy * rc[5] + gy * rc[8];

    // Sigma2d = JW * Sigma3d * JW^T + 0.3 I
    const float t00 = jw00 * c00 + jw01 * c01 + jw02 * c02;
    const float t01 = jw00 * c01 + jw01 * c11 + jw02 * c12;
    const float t02 = jw00 * c02 + jw01 * c12 + jw02 * c22;
    const float t10 = jw10 * c00 + jw11 * c01 + jw12 * c02;
    const float t11 = jw10 * c01 + jw11 * c11 + jw12 * c12;
    const float t12 = jw10 * c02 + jw11 * c12 + jw12 * c22;

    const float s00 = t00 * jw00 + t01 * jw01 + t02 * jw02 + 0.3f;
    const float s01 = t00 * jw10 + t01 * jw11 + t02 * jw12;
    const float s11 = t10 * jw10 + t11 * jw11 + t12 * jw12 + 0.3f;

    v2f m; m.x = mx; m.y = my;
    __builtin_nontemporal_store(m, ((v2f*)out_means) + i);
    v4f sg; sg.x = s00; sg.y = s01; sg.z = s01; sg.w = s11;
    __builtin_nontemporal_store(sg, ((v4f*)out_sigma) + i);
    __builtin_nontemporal_store(d, out_depth + i);
}

// ---- hot kernel: every block is a full 256-point tile ----
__global__ __launch_bounds__(PTS) void gs_project_main(
    const float* __restrict__ xyz,
    const float* __restrict__ log_scale,
    const v4f*  __restrict__ rotation,
    const float* __restrict__ R_cam,
    const float* __restrict__ t_cam,
    const float* __restrict__ intr,
    float* __restrict__ out_means,
    float* __restrict__ out_sigma,
    float* __restrict__ out_depth,
    int n)
{
    __shared__ float s_xyz[PTS * 3];
    __shared__ float s_ls [PTS * 3];

    const int tid  = threadIdx.x;
    const int base = blockIdx.x * PTS;
    const int i    = base + tid;

    // coalesced 128-bit async staging of the stride-3 arrays into LDS
    if (tid < (PTS * 3) / 4) {
        const v4f* gx = (const v4f*)(xyz       + (size_t)base * 3) + tid;
        const v4f* gl = (const v4f*)(log_scale + (size_t)base * 3) + tid;
#if HAVE_ASYNC_LDS
        __builtin_amdgcn_global_load_async_to_lds_b128(
            (as1_v4i*)gx, (as3_v4i*)(((v4f*)s_xyz) + tid), 0, 0);
        __builtin_amdgcn_global_load_async_to_lds_b128(
            (as1_v4i*)gl, (as3_v4i*)(((v4f*)s_ls) + tid), 0, 0);
#else
        ((v4f*)s_xyz)[tid] = __builtin_nontemporal_load(gx);
        ((v4f*)s_ls )[tid] = __builtin_nontemporal_load(gl);
#endif
    }
#if HAVE_ASYNC_LDS
#if __has_builtin(__builtin_amdgcn_s_wait_asynccnt)
    __builtin_amdgcn_s_wait_asynccnt(0);
#else
    asm volatile("s_wait_asynccnt 0" ::: "memory");
#endif
#endif
    __syncthreads();

    // stride-3 LDS reads: stride 3 coprime with 64 banks -> conflict-free
    const float px = s_xyz[3 * tid + 0];
    const float py = s_xyz[3 * tid + 1];
    const float pz = s_xyz[3 * tid + 2];
    const float l0 = s_ls [3 * tid + 0];
    const float l1 = s_ls [3 * tid + 1];
    const float l2 = s_ls [3 * tid + 2];

    // warm L2 for next block's rotation stream (global_prefetch_b8)
    if (i + PTS < n) __builtin_prefetch(&rotation[i + PTS], 0, 0);

    const v4f q = __builtin_nontemporal_load(&rotation[i]);

    gs_compute_store(i, px, py, pz, l0, l1, l2, q,
                     R_cam, t_cam, intr, out_means, out_sigma, out_depth);
}

// ---- tail kernel: < 256 leftover points, direct per-lane loads ----
__global__ __launch_bounds__(PTS) void gs_project_tail(
    const float* __restrict__ xyz,
    const float* __restrict__ log_scale,
    const v4f*  __restrict__ rotation,
    const float* __restrict__ R_cam,
    const float* __restrict__ t_cam,
    const float* __restrict__ intr,
    float* __restrict__ out_means,
    float* __restrict__ out_sigma,
    float* __restrict__ out_depth,
    int start, int n)
{
    const int i = start + threadIdx.x;
    if (i >= n) return;

    const float px = xyz[3 * (size_t)i + 0];
    const float py = xyz[3 * (size_t)i + 1];
    const float pz = xyz[3 * (size_t)i + 2];
    const float l0 = log_scale[3 * (size_t)i + 0];
    const float l1 = log_scale[3 * (size_t)i + 1];
    const float l2 = log_scale[3 * (size_t)i + 2];
    const v4f q = rotation[i];

    gs_compute_store(i, px, py, pz, l0, l1, l2, q,
                     R_cam, t_cam, intr, out_means, out_sigma, out_depth);
}

extern "C" void kernel_launch(void* const* d_in, const int* in_sizes, int n_in,
                              void* d_out, int out_size, void* d_ws, size_t ws_size,
                              hipStream_t stream) {
    const float* xyz  = (const float*)d_in[0];
    const float* ls   = (const float*)d_in[1];
    const v4f*   rot  = (const v4f*)  d_in[2];
    const float* Rc   = (const float*)d_in[3];
    const float* tc   = (const float*)d_in[4];
    const float* intr = (const float*)d_in[5];

    const int n = in_sizes[0] / 3;   // N points

    float* out       = (float*)d_out;
    float* out_means = out;                      // 2N floats
    float* out_sigma = out + (size_t)2 * n;      // 4N floats
    float* out_depth = out + (size_t)6 * n;      // N floats

    const int nfull = n / PTS;           // full 256-point tiles (15625 for N=4M)
    const int tail  = n - nfull * PTS;   // leftover points (0 for N=4M)

    if (nfull > 0) {
        hipLaunchKernelGGL(gs_project_main, dim3(nfull), dim3(PTS), 0, stream,
                           xyz, ls, rot, Rc, tc, intr,
                           out_means, out_sigma, out_depth, n);
    }
    if (tail > 0) {
        hipLaunchKernelGGL(gs_project_tail, dim3(1), dim3(PTS), 0, stream,
                           xyz, ls, rot, Rc, tc, intr,
                           out_means, out_sigma, out_depth, nfull * PTS, n);
    }
}